// GaussianFalloffKernelVelocity_84421877170796
// MI455X (gfx1250) — compile-verified
//
#include <hip/hip_runtime.h>

// CDNA5 / gfx1250: wave32, WMMA (not MFMA).
typedef __attribute__((ext_vector_type(2))) float v2f;
typedef __attribute__((ext_vector_type(4))) float v4f;
typedef __attribute__((ext_vector_type(8))) float v8f;

#define TWO_PI_F     6.2831853071795864769f
#define INV_TWO_PI_F 0.15915494309189533577f

// Shapes: vortex_feature (4,256,4) f32 [y,x,tau,sig]; points (4,128,128,2) f32;
// out velocity (4,128,128,2) f32.
//
// s[p,v] = tau_v*(1-exp(-r2/sig^2))/(2*pi*r2)   (nonlinear, VALU+trans)
//   vel.x[p] = p.x * rowsum(S) - (S @ vx)[p]
//   vel.y[p] = (S @ vy)[p]     - p.y * rowsum(S)
// Reduction over 256 vortices: D = S(16x256) x B(256x16), B cols [vx, vy, 1, 0..],
// accumulated with V_WMMA_F32_16X16X4_F32 (K-step 4). Each wave owns TWO 16-point
// tiles sharing one B fragment per K-step (2 WMMAs per B build / LDS load pair).
__global__ __launch_bounds__(256)
void vortex_velocity_wmma_kernel(const float* __restrict__ vf,
                                 const float* __restrict__ pts,
                                 float* __restrict__ out)
{
    __shared__ v4f   lds_vf[256];          // staged [vy, vx, tau/2pi, 1/sig^2] (4 KB)
    __shared__ float lds_c[8][2][3][16];   // per-wave, per-tile C-column scratch (3 KB)

    const int tid  = threadIdx.x;
    const int wave = tid >> 5;             // 0..7 (wave32)
    const int lane = tid & 31;
    const int hf   = lane >> 4;            // half-wave: K-pair select (A/B), row block (C)
    const int m    = lane & 15;            // A row (point), B column index

    // Wave gw handles tiles 2*gw and 2*gw+1 (16 points each). 512 waves per
    // batch, 8 waves per block => blocks never straddle a batch (512 % 8 == 0).
    const int gw    = blockIdx.x * 8 + wave;        // 0..2047
    const int b     = (blockIdx.x * 16) >> 10;      // uniform across block
    const int pbase = (gw & 511) * 32;              // first of 32 points in batch

    // Stage this batch's vortex features, pre-transformed so the hot loop has
    // no divides: z := tau/(2*pi), w := 1/sig^2 (one IEEE divide per vortex).
    {
        v4f raw = ((const v4f*)(vf + (size_t)b * 256 * 4))[tid];
        raw.z = raw.z * INV_TWO_PI_F;
        raw.w = 1.0f / (raw.w * raw.w);
        lds_vf[tid] = raw;
    }
    __syncthreads();

    // This lane's two points (row m of tile0 and tile1).
    const float* pp0 = pts + ((size_t)b * 16384 + (size_t)(pbase + m)) * 2;
    const float* pp1 = pp0 + 32;                    // +16 points
    const float py0 = pp0[0], px0 = pp0[1];
    const float py1 = pp1[0], px1 = pp1[1];

    // Per-lane B column selectors (col0 = vx, col1 = vy, col2 = rowsum).
    const float cx = (m == 0) ? 1.0f : 0.0f;
    const float cy = (m == 1) ? 1.0f : 0.0f;
    const float c1 = (m == 2) ? 1.0f : 0.0f;

    v8f c0 = {0.f, 0.f, 0.f, 0.f, 0.f, 0.f, 0.f, 0.f};
    v8f c1acc = {0.f, 0.f, 0.f, 0.f, 0.f, 0.f, 0.f, 0.f};

#pragma unroll 2
    for (int k = 0; k < 256; k += 4) {
        // A f32 16x4: element (M,K) at lane M+16*(K>>1), VGPR K&1.
        // This lane supplies strengths for vortices k0 = k+2*hf and k0+1.
        const int k0 = k + 2 * hf;
        const v4f f0 = lds_vf[k0];
        const v4f f1 = lds_vf[k0 + 1];

        // strength(point, vortex): ~10 VALU + exp + rcp, divide-free.
        auto strength = [](float py, float px, v4f f) -> float {
            const float dy = py - f.x;
            const float dx = px - f.y;
            const float r2 = fmaf(dy, dy, dx * dx);
            const float e  = __expf(-r2 * f.w);
            return (1.0f - e) * f.z * __builtin_amdgcn_rcpf(r2);
        };

        v2f a0, a1;
        a0.x = strength(py0, px0, f0);   // K even -> VGPR 0
        a0.y = strength(py0, px0, f1);   // K odd  -> VGPR 1
        a1.x = strength(py1, px1, f0);
        a1.y = strength(py1, px1, f1);

        // B f32 4x16 (mirror of A with N<->M): element (K,N) at lane
        // N+16*(K>>1), VGPR K&1. Shared by both tiles' WMMAs.
        v2f bb;
        bb.x = fmaf(cy, f0.x, fmaf(cx, f0.y, c1));
        bb.y = fmaf(cy, f1.x, fmaf(cx, f1.y, c1));

        // 8 args: (neg_a, A, neg_b, B, c_mod, C, reuse_a, reuse_b)
        c0    = __builtin_amdgcn_wmma_f32_16x16x4_f32(
                    false, a0, false, bb, (short)0, c0, false, false);
        c1acc = __builtin_amdgcn_wmma_f32_16x16x4_f32(
                    false, a1, false, bb, (short)0, c1acc, false, false);
    }

    // C layout (16x16 f32): lane L holds column N=L&15, rows M = j + 8*(L>>4)
    // in VGPR j. Publish the 3 useful columns (N=0: S@vx, N=1: S@vy, N=2: rowsum).
    if (m < 3) {
#pragma unroll
        for (int j = 0; j < 8; ++j) {
            lds_c[wave][0][m][hf * 8 + j] = c0[j];
            lds_c[wave][1][m][hf * 8 + j] = c1acc[j];
        }
    }
    __syncthreads();

    // Lane L emits component hf (0: x, 1: y) for point m of each tile.
    const float Sx0 = lds_c[wave][0][0][m];
    const float Sy0 = lds_c[wave][0][1][m];
    const float Ss0 = lds_c[wave][0][2][m];
    const float Sx1 = lds_c[wave][1][0][m];
    const float Sy1 = lds_c[wave][1][1][m];
    const float Ss1 = lds_c[wave][1][2][m];

    const float v0 = (hf == 0) ? (px0 * Ss0 - Sx0) : (Sy0 - py0 * Ss0);
    const float v1 = (hf == 0) ? (px1 * Ss1 - Sx1) : (Sy1 - py1 * Ss1);

    float* ob = out + ((size_t)b * 16384 + (size_t)(pbase + m)) * 2 + hf;
    ob[0]  = v0;
    ob[32] = v1;   // +16 points
}

extern "C" void kernel_launch(void* const* d_in, const int* in_sizes, int n_in,
                              void* d_out, int out_size, void* d_ws, size_t ws_size,
                              hipStream_t stream) {
    const float* vf  = (const float*)d_in[0];   // vortex_feature (4,256,4)
    const float* pts = (const float*)d_in[1];   // points (4,128,128,2)
    float* out = (float*)d_out;                 // velocity (4,128,128,2)

    // 4096 tiles of 16 points, 2 tiles per wave, 8 waves per 256-thread block.
    dim3 grid(256), block(256);
    hipLaunchKernelGGL(vortex_velocity_wmma_kernel, grid, block, 0, stream,
                       vf, pts, out);
}